// GNNEncoder_34041910788776
// MI455X (gfx1250) — compile-verified
//
#include <hip/hip_runtime.h>

#define N_NODES 50000
#define N_EDGES 800000
#define FEAT    128           // D == H == 128

typedef __attribute__((ext_vector_type(2))) float v2f;
typedef __attribute__((ext_vector_type(8))) float v8f;

// ---------------------------------------------------------------------------
// Zero a contiguous float span (agg + cnt live contiguously in workspace).
// ---------------------------------------------------------------------------
__global__ __launch_bounds__(256)
void zero_kernel(float4* __restrict__ p, int n4) {
    int i = blockIdx.x * blockDim.x + threadIdx.x;
    if (i < n4) p[i] = make_float4(0.f, 0.f, 0.f, 0.f);
}

// ---------------------------------------------------------------------------
// Edge scatter: one wave32 per edge. Lane l loads x[src][4l..4l+3] (float4,
// fully coalesced 512B per edge) and atomically accumulates into agg[dst].
// Lane 0 bumps the per-node edge count. x/agg are L2-resident (25.6MB each).
// ---------------------------------------------------------------------------
__global__ __launch_bounds__(256)
void scatter_kernel(const float* __restrict__ x,
                    const long long* __restrict__ src,
                    const long long* __restrict__ dst,
                    float* __restrict__ agg,
                    float* __restrict__ cnt) {
    int wave = (blockIdx.x * 256 + threadIdx.x) >> 5;   // 8 waves per block
    int lane = threadIdx.x & 31;
    if (wave >= N_EDGES) return;

    long long s = src[wave];
    long long d = dst[wave];

    const float4* xs = (const float4*)(x + (size_t)s * FEAT);
    float4 v = xs[lane];

    float* ad = agg + (size_t)d * FEAT + lane * 4;
    atomicAdd(ad + 0, v.x);
    atomicAdd(ad + 1, v.y);
    atomicAdd(ad + 2, v.z);
    atomicAdd(ad + 3, v.w);
    if (lane == 0) atomicAdd(&cnt[d], 1.0f);
}

// ---------------------------------------------------------------------------
// Fused SAGE layer: out = relu( l2norm( mean @ Wl + bl + x @ Wr ) )
// One workgroup = 16 output rows x 128 cols. 8 waves, one 16x16 col tile per
// wave. A = [mean | x] (16x256 f32) staged in LDS (stride 257: bank-conflict
// free). K walked in steps of 4 with V_WMMA_F32_16X16X4_F32 (64 WMMAs/wave).
// B fragments come straight from global (weights are L2/WGP$ resident).
// ---------------------------------------------------------------------------
__global__ __launch_bounds__(256)
void sage_gemm_kernel(const float* __restrict__ xin,
                      const float* __restrict__ agg,
                      const float* __restrict__ cnt,
                      const float* __restrict__ Wl,
                      const float* __restrict__ bl,
                      const float* __restrict__ Wr,
                      float* __restrict__ out) {
    __shared__ float Alds[16 * 257];             // [m][k], k in 0..255
    __shared__ __align__(16) float Olds[16 * 128];
    __shared__ float RowInv[16];                 // 1/max(cnt,1)
    __shared__ float Psum[16 * 16];
    __shared__ float NormInv[16];

    const int tid  = threadIdx.x;
    const int row0 = blockIdx.x * 16;

    if (tid < 16) {
        float c = cnt[row0 + tid];
        RowInv[tid] = 1.0f / fmaxf(c, 1.0f);
    }
    __syncthreads();

    // Stage A tile: cols 0..127 = mean (agg * RowInv), cols 128..255 = x.
    for (int i = tid; i < 16 * 256; i += 256) {
        int m = i >> 8;
        int k = i & 255;
        int r = row0 + m;
        float v;
        if (k < 128) v = agg[(size_t)r * FEAT + k] * RowInv[m];
        else         v = xin[(size_t)r * FEAT + (k - 128)];
        Alds[m * 257 + k] = v;
    }
    __syncthreads();

    const int wave  = tid >> 5;
    const int lane  = tid & 31;
    const int n0    = wave * 16;          // this wave's 16-column tile
    const int m     = lane & 15;          // A-matrix row (16x4 f32 layout)
    const int khalf = (lane >> 4) * 2;    // lanes 16-31 hold K+2, K+3
    const int n     = lane & 15;          // B/C column

    v8f acc = {};

    // mean @ Wl  (A cols 0..127)
    for (int kk = 0; kk < 128; kk += 4) {
        int k0 = kk + khalf;
        v2f a, b;
        a.x = Alds[m * 257 + k0];
        a.y = Alds[m * 257 + k0 + 1];
        b.x = Wl[(size_t)k0 * FEAT + n0 + n];
        b.y = Wl[(size_t)(k0 + 1) * FEAT + n0 + n];
        acc = __builtin_amdgcn_wmma_f32_16x16x4_f32(
            false, a, false, b, (short)0, acc, false, false);
    }
    // x @ Wr  (A cols 128..255)
    for (int kk = 0; kk < 128; kk += 4) {
        int k0 = kk + khalf;
        v2f a, b;
        a.x = Alds[m * 257 + 128 + k0];
        a.y = Alds[m * 257 + 128 + k0 + 1];
        b.x = Wr[(size_t)k0 * FEAT + n0 + n];
        b.y = Wr[(size_t)(k0 + 1) * FEAT + n0 + n];
        acc = __builtin_amdgcn_wmma_f32_16x16x4_f32(
            false, a, false, b, (short)0, acc, false, false);
    }

    // Add bias, spill 16x16 tile to LDS (C/D layout: VGPR r -> M=r / M=r+8).
    float bias = bl[n0 + n];
    int mbase = (lane >> 4) * 8;
#pragma unroll
    for (int r = 0; r < 8; r++) {
        Olds[(mbase + r) * 128 + n0 + n] = acc[r] + bias;
    }
    __syncthreads();

    // Row-wise sum of squares across all 128 cols (cross-wave via LDS).
    {
        int rr = tid >> 4;
        int cc = tid & 15;
        float s = 0.f;
#pragma unroll
        for (int j = 0; j < 8; j++) {
            float v = Olds[rr * 128 + cc + j * 16];
            s += v * v;
        }
        Psum[rr * 16 + cc] = s;
    }
    __syncthreads();
    if (tid < 16) {
        float s = 0.f;
#pragma unroll
        for (int j = 0; j < 16; j++) s += Psum[tid * 16 + j];
        NormInv[tid] = 1.0f / fmaxf(sqrtf(s), 1e-12f);
    }
    __syncthreads();

    // relu(out * 1/norm) -> global, float4 stores (2048 floats / block).
    for (int i = tid; i < 16 * 32; i += 256) {
        int mm = i >> 5;
        int c4 = i & 31;
        float4 v = ((const float4*)Olds)[mm * 32 + c4];
        float ninv = NormInv[mm];
        v.x = fmaxf(v.x * ninv, 0.f);
        v.y = fmaxf(v.y * ninv, 0.f);
        v.z = fmaxf(v.z * ninv, 0.f);
        v.w = fmaxf(v.w * ninv, 0.f);
        ((float4*)(out + (size_t)(row0 + mm) * FEAT))[c4] = v;
    }
}

// ---------------------------------------------------------------------------
extern "C" void kernel_launch(void* const* d_in, const int* in_sizes, int n_in,
                              void* d_out, int out_size, void* d_ws, size_t ws_size,
                              hipStream_t stream) {
    const float*     x   = (const float*)d_in[0];
    const long long* ei  = (const long long*)d_in[1];   // int64 edge_index (2, E)
    const float*     W1l = (const float*)d_in[2];
    const float*     b1l = (const float*)d_in[3];
    const float*     W1r = (const float*)d_in[4];
    const float*     W2l = (const float*)d_in[5];
    const float*     b2l = (const float*)d_in[6];
    const float*     W2r = (const float*)d_in[7];
    float*           out = (float*)d_out;

    const long long* src = ei;
    const long long* dst = ei + N_EDGES;

    // Workspace layout (contiguous): agg | cnt | h1
    const size_t aggFloats = (size_t)N_NODES * FEAT;    // 6,400,000
    const size_t cntFloats = (size_t)N_NODES;           //    50,000
    char* ws = (char*)d_ws;
    float* agg = (float*)ws;
    float* cnt = (float*)(ws + aggFloats * sizeof(float));
    float* h1  = (float*)(ws + (aggFloats + cntFloats) * sizeof(float));

    const int zeroN4   = (int)((aggFloats + cntFloats) / 4);       // 1,612,500
    const int zeroGrid = (zeroN4 + 255) / 256;
    const int scatGrid = N_EDGES / 8;                              // 8 waves/block
    const int gemmGrid = N_NODES / 16;                             // 3125 exact

    // ---- layer 1 ----
    zero_kernel<<<zeroGrid, 256, 0, stream>>>((float4*)agg, zeroN4);
    scatter_kernel<<<scatGrid, 256, 0, stream>>>(x, src, dst, agg, cnt);
    sage_gemm_kernel<<<gemmGrid, 256, 0, stream>>>(x, agg, cnt, W1l, b1l, W1r, h1);

    // ---- layer 2 ----
    zero_kernel<<<zeroGrid, 256, 0, stream>>>((float4*)agg, zeroN4);
    scatter_kernel<<<scatGrid, 256, 0, stream>>>(h1, src, dst, agg, cnt);
    sage_gemm_kernel<<<gemmGrid, 256, 0, stream>>>(h1, agg, cnt, W2l, b2l, W2r, out);
}